// GraphSAGE_420906795016
// MI455X (gfx1250) — compile-verified
//
#include <hip/hip_runtime.h>
#include <hip/hip_bf16.h>
#include <stdint.h>

// GraphSAGE (2-layer, mean aggr) for MI455X / gfx1250.
//
// Cost model: edge scatter dominates (~5GB of L2-resident gather/scatter,
// ~600M fp32 atomics); GEMMs are ~17 GFLOP -> microseconds. So: simple
// atomic scatter at full L2 rate, fp32 WMMA (v_wmma_f32_16x16x4_f32) for the
// linear layers (exact fp32 math), 1/deg folded into the WMMA A-load, and
// weight column-tiles staged once per block into LDS via the gfx1250 async
// load-to-LDS path (ASYNCcnt) so B fragments come from ds_load instead of
// being re-fetched from L2 by every M-tile wave.

#define N_NODES  100000
#define N_EDGES  1600000
#define IN_FEATS 128
#define HIDDEN   256
#define CLASSES  41

typedef float v2f __attribute__((ext_vector_type(2)));
typedef float v8f __attribute__((ext_vector_type(8)));

// ---------------------------------------------------------------------------
// gfx1250 async global->LDS copy (tracked by ASYNCcnt, no VGPR round-trip).
// Builtin signature (from hipcc diagnostic): arg0 is a global (AS1) int*.
// AS3 pointer representation: low 32 bits of a flat shared address are the
// LDS byte offset (flat->LDS truncates), so integer round-trip casts are safe.
// ---------------------------------------------------------------------------
__device__ __forceinline__ void async_ld_b32(const float* g, float* l) {
#if __has_builtin(__builtin_amdgcn_global_load_async_to_lds_b32)
  typedef __attribute__((address_space(1))) int gi_t;
  typedef __attribute__((address_space(3))) int li_t;
  __builtin_amdgcn_global_load_async_to_lds_b32(
      (gi_t*)(uintptr_t)g, (li_t*)(uintptr_t)l, 0, 0);
#else
  unsigned lds_off = (unsigned)(uintptr_t)l;
  asm volatile("global_load_async_to_lds_b32 %0, %1, off"
               :: "v"(lds_off), "v"(g) : "memory");
#endif
}

__device__ __forceinline__ void wait_async0() {
#if __has_builtin(__builtin_amdgcn_s_wait_asynccnt)
  __builtin_amdgcn_s_wait_asynccnt(0);
#else
  asm volatile("s_wait_asynccnt 0x0" ::: "memory");
#endif
}

// ---------------------------------------------------------------------------
// Layer-1 scatter: one wave per edge. 32 lanes x float4 = the whole 128-float
// src row, fully coalesced (512B per wave). Lane 0 also bumps the degree.
// ---------------------------------------------------------------------------
__global__ void sage_scatter1(const float* __restrict__ x,
                              const int* __restrict__ ei,
                              float* __restrict__ agg,   // [N,128] pre-zeroed
                              float* __restrict__ deg) { // [N] pre-zeroed
  long long gid = (long long)blockIdx.x * blockDim.x + threadIdx.x;
  int e    = (int)(gid >> 5);
  int lane = (int)(gid & 31);
  if (e >= N_EDGES) return;
  int s = ei[e];
  int d = ei[N_EDGES + e];
  const float4 v = *(const float4*)(x + (size_t)s * IN_FEATS + lane * 4);
  float* p = agg + (size_t)d * IN_FEATS + lane * 4;
  atomicAdd(p + 0, v.x);
  atomicAdd(p + 1, v.y);
  atomicAdd(p + 2, v.z);
  atomicAdd(p + 3, v.w);
  if (lane == 0) atomicAdd(deg + d, 1.0f);
}

// ---------------------------------------------------------------------------
// Layer-2 scatter: 64 threads (float4 each) per edge over the 256-float row.
// ---------------------------------------------------------------------------
__global__ void sage_scatter2(const float* __restrict__ h,
                              const int* __restrict__ ei,
                              float* __restrict__ agg) { // [N,256] pre-zeroed
  long long gid = (long long)blockIdx.x * blockDim.x + threadIdx.x;
  int e  = (int)(gid >> 6);
  int fb = (int)(gid & 63) * 4;
  if (e >= N_EDGES) return;
  int s = ei[e];
  int d = ei[N_EDGES + e];
  const float4 v = *(const float4*)(h + (size_t)s * HIDDEN + fb);
  float* p = agg + (size_t)d * HIDDEN + fb;
  atomicAdd(p + 0, v.x);
  atomicAdd(p + 1, v.y);
  atomicAdd(p + 2, v.z);
  atomicAdd(p + 3, v.w);
}

// deg -> 1/max(deg,1), in place (matches jnp.clip(deg, 1.0))
__global__ void sage_rdeg(float* __restrict__ deg) {
  int i = blockIdx.x * blockDim.x + threadIdx.x;
  if (i < N_NODES) deg[i] = 1.0f / fmaxf(deg[i], 1.0f);
}

// ---------------------------------------------------------------------------
// Fused SAGE linear: out = act( (Aagg*rdeg) @ Wl + Aself @ Wr + bias )
// Block = 8 waves = 8 consecutive M-tiles, all sharing one 16-col N-tile
// (blockIdx.y). Weight column-tiles for the full K are async-DMA'd to LDS
// once per block; padded columns are zero-filled with ds_store (disjoint
// addresses, so no DS<->ASYNC ordering hazard). Each wave then runs the
// V_WMMA_F32_16X16X4_F32 K-loop with A from global, B from LDS.
//   A 16x4 f32 frag : lane r=lane&15 -> row m0+r; VGPR v -> K = k0+2*half+v
//   B  4x16 f32 frag: symmetric (VGPR v -> K row k0+2*half+v, col n0+r)
//   C/D 16x16       : acc[v] -> row m0+v+8*half, col n0+r
// ---------------------------------------------------------------------------
__global__ void sage_gemm(const float* __restrict__ Aagg,
                          const float* __restrict__ Aself,
                          const float* __restrict__ Wl,   // [K, NC] row-major
                          const float* __restrict__ Wr,   // [K, NC] row-major
                          const float* __restrict__ bias, // [NC]
                          const float* __restrict__ rdeg, // [N]
                          float* __restrict__ out,        // [N, NC]
                          int K, int NC, int relu) {
  __shared__ float sB[2 * 256 * 16];  // [Wl | Wr] tiles, layout [k*16+c]; 32KB

  const int n0 = blockIdx.y * 16;
  const int tileElems = K * 16;

  // ---- stage both weight column-tiles into LDS (async DMA) ----
  for (int idx = threadIdx.x; idx < 2 * tileElems; idx += 256) {
    int mtx = idx >= tileElems;
    int rem = mtx ? idx - tileElems : idx;
    int k = rem >> 4;
    int c = rem & 15;
    int col = n0 + c;
    const float* W = mtx ? Wr : Wl;
    if (col < NC)
      async_ld_b32(W + (size_t)k * NC + col, &sB[idx]);
    else
      sB[idx] = 0.0f;               // pad column -> zero (ds_store)
  }
  wait_async0();                    // this wave's async transfers landed in LDS
  __syncthreads();                  // all waves' tiles visible

  int mtile = blockIdx.x * 8 + (threadIdx.x >> 5);
  if (mtile >= N_NODES / 16) return;   // wave-uniform exit (EXEC stays all-1s)

  int lane = threadIdx.x & 31;
  int half = lane >> 4;             // 0: K pair {0,1}, 1: K pair {2,3}
  int r    = lane & 15;
  int m0   = mtile * 16;
  int c    = n0 + r;

  float scale = rdeg[m0 + r];
  const float* a0 = Aagg  + (size_t)(m0 + r) * K;
  const float* a1 = Aself + (size_t)(m0 + r) * K;
  const float* bl = sB;
  const float* br = sB + tileElems;

  v8f acc = {};
  // Phase 1: (agg/deg) @ Wl
  for (int k0 = 0; k0 < K; k0 += 4) {
    int ka = k0 + 2 * half;
    v2f a, b;
    a.x = a0[ka]     * scale;
    a.y = a0[ka + 1] * scale;
    b.x = bl[ka * 16 + r];
    b.y = bl[(ka + 1) * 16 + r];
    acc = __builtin_amdgcn_wmma_f32_16x16x4_f32(false, a, false, b,
                                                (short)0, acc, false, false);
  }
  // Phase 2: h @ Wr
  for (int k0 = 0; k0 < K; k0 += 4) {
    int ka = k0 + 2 * half;
    v2f a, b;
    a.x = a1[ka];
    a.y = a1[ka + 1];
    b.x = br[ka * 16 + r];
    b.y = br[(ka + 1) * 16 + r];
    acc = __builtin_amdgcn_wmma_f32_16x16x4_f32(false, a, false, b,
                                                (short)0, acc, false, false);
  }

  // Epilogue: bias (+ReLU), masked store of the 16x16 tile
  if (c < NC) {
    float bv = bias[c];
#pragma unroll
    for (int v = 0; v < 8; ++v) {
      float val = acc[v] + bv;
      if (relu) val = fmaxf(val, 0.0f);
      out[(size_t)(m0 + v + 8 * half) * NC + c] = val;
    }
  }
}

// ---------------------------------------------------------------------------
// In-place log_softmax over 41 classes: one wave (32 lanes) per row,
// shfl_xor tree reductions (wave32).
// ---------------------------------------------------------------------------
__global__ void sage_log_softmax(float* __restrict__ out) {
  int wave = (int)(((long long)blockIdx.x * blockDim.x + threadIdx.x) >> 5);
  int lane = threadIdx.x & 31;
  if (wave >= N_NODES) return;
  float* row = out + (size_t)wave * CLASSES;

  const float NEG = -3.402823466e+38f;
  float v0 = (lane < CLASSES)      ? row[lane]      : NEG;
  float v1 = (lane + 32 < CLASSES) ? row[lane + 32] : NEG;

  float m = fmaxf(v0, v1);
  for (int off = 16; off > 0; off >>= 1)
    m = fmaxf(m, __shfl_xor(m, off, 32));

  float s = 0.0f;
  if (lane < CLASSES)      s += expf(v0 - m);
  if (lane + 32 < CLASSES) s += expf(v1 - m);
  for (int off = 16; off > 0; off >>= 1)
    s += __shfl_xor(s, off, 32);

  float lse = m + logf(s);
  if (lane < CLASSES)      row[lane]      = v0 - lse;
  if (lane + 32 < CLASSES) row[lane + 32] = v1 - lse;
}

// ---------------------------------------------------------------------------
extern "C" void kernel_launch(void* const* d_in, const int* in_sizes, int n_in,
                              void* d_out, int out_size, void* d_ws, size_t ws_size,
                              hipStream_t stream) {
  const float* x   = (const float*)d_in[0];
  const int*   ei  = (const int*)  d_in[1];  // [2,E] int32
  const float* W1l = (const float*)d_in[2];
  const float* W1r = (const float*)d_in[3];
  const float* b1  = (const float*)d_in[4];
  const float* W2l = (const float*)d_in[5];
  const float* W2r = (const float*)d_in[6];
  const float* b2  = (const float*)d_in[7];
  float* out = (float*)d_out;

  // Workspace layout (~206MB total):
  //   [0,1MB)        deg -> rdeg (in place)
  //   [1MB, +102.4M) bufA: agg1 (stride 128), reused as agg2 (stride 256)
  //   [.., +102.4M)  bufB: h1 (stride 256)
  char* ws = (char*)d_ws;
  const size_t HDR = (size_t)1 << 20;
  const size_t BUF = (size_t)N_NODES * HIDDEN * sizeof(float);
  float* deg  = (float*)ws;
  float* bufA = (float*)(ws + HDR);
  float* bufB = (float*)(ws + HDR + BUF);

  // ---- Layer 1 ----
  (void)hipMemsetAsync(ws, 0, HDR + (size_t)N_NODES * IN_FEATS * sizeof(float), stream);
  {
    long long th = (long long)N_EDGES * 32;
    sage_scatter1<<<(int)((th + 255) / 256), 256, 0, stream>>>(x, ei, bufA, deg);
  }
  sage_rdeg<<<(N_NODES + 255) / 256, 256, 0, stream>>>(deg);
  {
    dim3 grid((N_NODES / 16 + 7) / 8, HIDDEN / 16);  // 782 x 16
    sage_gemm<<<grid, 256, 0, stream>>>(
        bufA, x, W1l, W1r, b1, deg, bufB, IN_FEATS, HIDDEN, /*relu=*/1);
  }

  // ---- Layer 2 ----
  (void)hipMemsetAsync(bufA, 0, BUF, stream);
  {
    long long th = (long long)N_EDGES * (HIDDEN / 4);
    sage_scatter2<<<(int)((th + 255) / 256), 256, 0, stream>>>(bufB, ei, bufA);
  }
  {
    dim3 grid((N_NODES / 16 + 7) / 8, (CLASSES + 15) / 16);  // 782 x 3
    sage_gemm<<<grid, 256, 0, stream>>>(
        bufA, bufB, W2l, W2r, b2, deg, out, HIDDEN, CLASSES, /*relu=*/0);
  }

  // ---- log_softmax (in place on d_out) ----
  sage_log_softmax<<<(N_NODES + 7) / 8, 256, 0, stream>>>(out);
}